// BLInputLayer_74594991997074
// MI455X (gfx1250) — compile-verified
//
#include <hip/hip_runtime.h>
#include <hip/hip_bf16.h>
#include <stdint.h>

typedef __attribute__((ext_vector_type(2))) float v2f;
typedef __attribute__((ext_vector_type(8))) float v8f;

#define RADIX      256
#define RBITS      8
#define SCAN_BLOCK 256
#define SCAN_ITEMS 4
#define SCAN_TILE  (SCAN_BLOCK * SCAN_ITEMS)   // 1024

// ---------------------------------------------------------------- key build
__global__ void k_keys(const int* __restrict__ coords, uint32_t* __restrict__ keys,
                       uint32_t* __restrict__ idx, int n, int L, int S) {
  int i = blockIdx.x * blockDim.x + threadIdx.x;
  if (i >= n) return;
  uint32_t b = (uint32_t)(i / L);
  uint32_t x = (uint32_t)coords[i * 3 + 0];
  uint32_t y = (uint32_t)coords[i * 3 + 1];
  uint32_t z = (uint32_t)coords[i * 3 + 2];
  uint32_t k = ((b * (uint32_t)S + x) * (uint32_t)S + y) * (uint32_t)S + z;
  keys[i] = k;
  idx[i]  = (uint32_t)i;
}

// ------------------------------------------------------------- radix: count
__global__ void k_radix_count(const uint32_t* __restrict__ keys,
                              uint32_t* __restrict__ blockHist,
                              int n, int shift, int NB) {
  __shared__ uint32_t hist[RADIX];
  int t = threadIdx.x;
  hist[t] = 0u;
  __syncthreads();
  int i = blockIdx.x * 256 + t;
  if (i < n) atomicAdd(&hist[(keys[i] >> shift) & 0xFFu], 1u);
  __syncthreads();
  // digit-major layout: exclusive scan of this array gives stable global bases
  blockHist[(uint32_t)t * NB + blockIdx.x] = hist[t];
}

// ----------------------------------------------------------- radix: scatter
__global__ void k_radix_scatter(const uint32_t* __restrict__ keysIn,
                                const uint32_t* __restrict__ idxIn,
                                uint32_t* __restrict__ keysOut,
                                uint32_t* __restrict__ idxOut,
                                const uint32_t* __restrict__ base,
                                int n, int shift, int NB) {
  __shared__ uint32_t whist[8 * RADIX];
  int t = threadIdx.x, lane = t & 31, wid = t >> 5;
  for (int j = t; j < 8 * RADIX; j += 256) whist[j] = 0u;
  __syncthreads();

  int i = blockIdx.x * 256 + t;
  bool valid = (i < n);
  uint32_t key = valid ? keysIn[i] : 0u;
  uint32_t d   = (key >> shift) & 0xFFu;

  // wave32 peer matching: deterministic stable rank within wave
  unsigned peers = 0xffffffffu;
  #pragma unroll
  for (int b = 0; b < RBITS; ++b) {
    unsigned bit = (d >> b) & 1u;
    unsigned bal = (unsigned)__ballot(bit ? 1 : 0);
    peers &= bit ? bal : ~bal;
  }
  peers &= (unsigned)__ballot(valid ? 1 : 0);
  unsigned lt  = (lane == 0) ? 0u : (0xffffffffu >> (32 - lane));
  unsigned rIW = (unsigned)__popc(peers & lt);
  if (valid && rIW == 0u) whist[wid * RADIX + d] = (uint32_t)__popc(peers);
  __syncthreads();

  if (valid) {
    uint32_t off = 0u;
    for (int w = 0; w < wid; ++w) off += whist[w * RADIX + d];
    uint32_t dst = base[d * (uint32_t)NB + blockIdx.x] + off + rIW;
    keysOut[dst] = key;
    idxOut[dst]  = idxIn[i];
  }
}

// -------------------------------------------------------- exclusive scan u32
__global__ void k_scan_ex(const uint32_t* __restrict__ in, uint32_t* __restrict__ out,
                          uint32_t* __restrict__ sums, int m) {
  __shared__ uint32_t lds[SCAN_BLOCK];
  int t = threadIdx.x;
  int base = blockIdx.x * SCAN_TILE + t * SCAN_ITEMS;
  uint32_t v[SCAN_ITEMS];
  uint32_t tsum = 0u;
  #pragma unroll
  for (int j = 0; j < SCAN_ITEMS; ++j) { v[j] = (base + j < m) ? in[base + j] : 0u; tsum += v[j]; }
  lds[t] = tsum;
  __syncthreads();
  for (int off = 1; off < SCAN_BLOCK; off <<= 1) {
    uint32_t x = (t >= off) ? lds[t - off] : 0u;
    __syncthreads();
    lds[t] += x;
    __syncthreads();
  }
  uint32_t ex = lds[t] - tsum;
  if (t == SCAN_BLOCK - 1) sums[blockIdx.x] = lds[t];
  #pragma unroll
  for (int j = 0; j < SCAN_ITEMS; ++j) { if (base + j < m) out[base + j] = ex; ex += v[j]; }
}

__global__ void k_scan_add(uint32_t* __restrict__ out, const uint32_t* __restrict__ sums, int m) {
  int i = blockIdx.x * blockDim.x + threadIdx.x;
  if (i < m) out[i] += sums[i / SCAN_TILE];
}

// ---------------------------------------------------------- heads and ranks
__global__ void k_heads(const uint32_t* __restrict__ keys, uint32_t* __restrict__ flags, int n) {
  int i = blockIdx.x * blockDim.x + threadIdx.x;
  if (i >= n) return;
  flags[i] = (i == 0 || keys[i] != keys[i - 1]) ? 1u : 0u;
}

__global__ void k_rank_fix(uint32_t* __restrict__ ranks, const uint32_t* __restrict__ flags, int n) {
  int i = blockIdx.x * blockDim.x + threadIdx.x;
  if (i < n) ranks[i] = ranks[i] + flags[i] - 1u;  // inclusive-scan-of-flags minus 1
}

// ------------------------------------------------------------------ zeroing
__global__ void k_zero4(float4* __restrict__ out4, long long m4,
                        float* __restrict__ out, long long m) {
  long long i = (long long)blockIdx.x * blockDim.x + threadIdx.x;
  long long stride = (long long)gridDim.x * blockDim.x;
  float4 z; z.x = 0.f; z.y = 0.f; z.z = 0.f; z.w = 0.f;
  for (; i < m4; i += stride) out4[i] = z;            // global_store_b128
  if (blockIdx.x == 0 && threadIdx.x == 0)
    for (long long j = m4 * 4; j < m; ++j) out[j] = 0.f;
}

// --------------------------------------------- WMMA-based segment scatter-sum
// Per 16 sorted rows: Out(16xC) = M(16x16) x F(16xC), M one-hot from local ranks.
// f32 WMMA 16x16x4: A 16x4 (2 VGPR), B 4x16 (2 VGPR), C/D 16x16 (8 VGPR).
// Branchless: EXEC stays all-ones everywhere (WMMA requirement); tail rows are
// masked to zero inside the A matrix, so clamped addresses only ever add 0.0.
__global__ void k_wmma_scatter(const float* __restrict__ feats,
                               const uint32_t* __restrict__ orig,
                               const uint32_t* __restrict__ ranks,
                               float* __restrict__ out, int n, int C) {
  int lane = threadIdx.x & 31;
  int wid  = threadIdx.x >> 5;
  long long tile = (long long)blockIdx.x * (blockDim.x >> 5) + wid;
  long long ts   = tile * 16;
  if (ts >= n) return;                       // uniform per wave

  int m  = lane & 15;                        // M row / N col index for this lane
  int hi = lane >> 4;                        // lane half selects K sub-slice

  int rowIdx = (int)ts + m;
  if (rowIdx >= n) rowIdx = n - 1;           // tail clamp (no divergence)
  uint32_t myRank = ranks[rowIdx];
  uint32_t myOrig = orig[rowIdx];
  uint32_t R0 = __shfl(myRank, 0);

  // A chunks + hoisted B-row pointers.
  // chunk c covers K = 4c..4c+3; VGPR .x <-> K = 4c + 2*hi, .y <-> K = 4c+1+2*hi
  v2f A[4];
  const float* __restrict__ P0[4];
  const float* __restrict__ P1[4];
  #pragma unroll
  for (int c = 0; c < 4; ++c) {
    int j0 = 4 * c + 0 + 2 * hi;
    int j1 = 4 * c + 1 + 2 * hi;
    uint32_t rl0 = __shfl(myRank, j0) - R0;
    uint32_t rl1 = __shfl(myRank, j1) - R0;
    bool v0 = (ts + j0) < n;                 // zero A for tail rows -> exact
    bool v1 = (ts + j1) < n;
    A[c].x = (v0 && rl0 == (uint32_t)m) ? 1.0f : 0.0f;
    A[c].y = (v1 && rl1 == (uint32_t)m) ? 1.0f : 0.0f;
    P0[c] = feats + (long long)__shfl(myOrig, j0) * C;
    P1[c] = feats + (long long)__shfl(myOrig, j1) * C;
  }
  __builtin_prefetch(P0[0] + m, 0, 0);       // global_prefetch of gathered rows
  __builtin_prefetch(P1[2] + m, 0, 0);

  // hoisted output row bases (clamped; clamped rows only receive +0.0)
  float* __restrict__ RB[8];
  #pragma unroll
  for (int r = 0; r < 8; ++r) {
    long long outRow = (long long)R0 + (r + 8 * hi);
    if (outRow >= n) outRow = n - 1;         // address clamp, value is 0 there
    RB[r] = out + outRow * (long long)C;
  }

  int ntiles = C >> 4;
  for (int nt = 0; nt < ntiles; ++nt) {
    int col = nt * 16 + m;
    v8f acc = {};
    #pragma unroll
    for (int c = 0; c < 4; ++c) {
      v2f Bv;
      Bv.x = P0[c][col];
      Bv.y = P1[c][col];
      acc = __builtin_amdgcn_wmma_f32_16x16x4_f32(
          /*neg_a=*/false, A[c], /*neg_b=*/false, Bv,
          /*c_mod=*/(short)0, acc, /*reuse_a=*/false, /*reuse_b=*/false);
    }
    #pragma unroll
    for (int r = 0; r < 8; ++r)
      atomicAdd(RB[r] + col, acc[r]);        // branchless global_atomic_add_f32
  }
}

// ---------------------------------------------------------------- launcher
extern "C" void kernel_launch(void* const* d_in, const int* in_sizes, int n_in,
                              void* d_out, int out_size, void* d_ws, size_t ws_size,
                              hipStream_t stream) {
  const int*   coords = (const int*)d_in[0];
  const float* feats  = (const float*)d_in[1];
  float*       out    = (float*)d_out;

  int n = in_sizes[0] / 3;          // B*L = 262144
  int C = in_sizes[1] / n;          // 128
  int L = n / 8;                    // B = 8 (reference constant)
  int S = 512;                      // reference constant

  // workspace carve (256B aligned)
  uint8_t* w = (uint8_t*)d_ws;
  auto carve = [&](size_t bytes) -> uint32_t* {
    uint32_t* p = (uint32_t*)w;
    w += (bytes + 255) & ~(size_t)255;
    return p;
  };
  uint32_t* keysA = carve((size_t)n * 4);
  uint32_t* keysB = carve((size_t)n * 4);
  uint32_t* idxA  = carve((size_t)n * 4);
  uint32_t* idxB  = carve((size_t)n * 4);
  uint32_t* flags = carve((size_t)n * 4);
  uint32_t* ranks = carve((size_t)n * 4);
  int NB = (n + 255) / 256;
  uint32_t* bh    = carve((size_t)RADIX * NB * 4);
  uint32_t* sums  = carve(2048 * 4);
  uint32_t* sums2 = carve(64 * 4);

  int gp = (n + 255) / 256;

  auto scan_ex = [&](uint32_t* in, uint32_t* outb, int m) {
    int g = (m + SCAN_TILE - 1) / SCAN_TILE;           // <= 1024 for our sizes
    k_scan_ex<<<g, SCAN_BLOCK, 0, stream>>>(in, outb, sums, m);
    k_scan_ex<<<1, SCAN_BLOCK, 0, stream>>>(sums, sums, sums2, g);
    k_scan_add<<<(m + 255) / 256, 256, 0, stream>>>(outb, sums, m);
  };

  // 1) keys
  k_keys<<<gp, 256, 0, stream>>>(coords, keysA, idxA, n, L, S);

  // 2) 4-pass LSD radix sort (8-bit digits, stable)
  uint32_t *ki = keysA, *ii = idxA, *ko = keysB, *io = idxB;
  for (int pass = 0; pass < 4; ++pass) {
    int shift = pass * 8;
    k_radix_count<<<NB, 256, 0, stream>>>(ki, bh, n, shift, NB);
    scan_ex(bh, bh, RADIX * NB);
    k_radix_scatter<<<NB, 256, 0, stream>>>(ki, ii, ko, io, bh, n, shift, NB);
    uint32_t* t;
    t = ki; ki = ko; ko = t;
    t = ii; ii = io; io = t;
  }
  // after 4 passes sorted data is back in keysA/idxA (== ki/ii)

  // 3) head flags -> ranks (inclusive scan of flags, minus 1)
  k_heads<<<gp, 256, 0, stream>>>(ki, flags, n);
  scan_ex(flags, ranks, n);
  k_rank_fix<<<gp, 256, 0, stream>>>(ranks, flags, n);

  // 4) zero output (padded slots must be zero; harness poisons d_out)
  long long m  = (long long)out_size;
  long long m4 = m >> 2;
  k_zero4<<<4096, 256, 0, stream>>>((float4*)out, m4, out, m);

  // 5) WMMA selection-matrix segment-sum scatter
  long long tiles  = ((long long)n + 15) / 16;
  int       blocks = (int)((tiles + 7) / 8);           // 8 waves (tiles) per 256-thread block
  k_wmma_scatter<<<blocks, 256, 0, stream>>>(feats, ii, ranks, out, n, C);
}